// WeightQuantizer_43765716746470
// MI455X (gfx1250) — compile-verified
//
#include <hip/hip_runtime.h>
#include <math.h>

// WeightQuantizer: per-128-group min/max quantize + soft-round (sum of tanh steps).
// x: (2048,4096) fp32 -> out: (2048,4096) fp32.
// One wave32 per group of 128: lane l owns elements [4l, 4l+4) via b128 loads.
//
// Key CDNA5 paths: hardware v_tanh_f32 (TRANS, co-executes with VALU),
// global_load_b128/global_store_b128 streaming, wave32 ds-shuffle reduction.

#define GROUP_SIZE 128
#define QMAX       15
#define T_SHARP    1000.0f
#define CLIPMIN    1e-6f

// gfx1250 hardware tanh (TRANS op). Fall back to libm if builtin unavailable.
#if __has_builtin(__builtin_amdgcn_tanhf)
__device__ __forceinline__ float dev_tanh(float x) { return __builtin_amdgcn_tanhf(x); }
#elif __has_builtin(__builtin_amdgcn_tanh_f32)
__device__ __forceinline__ float dev_tanh(float x) { return __builtin_amdgcn_tanh_f32(x); }
#else
__device__ __forceinline__ float dev_tanh(float x) { return tanhf(x); }
#endif

__global__ __launch_bounds__(256) void WeightQuantizer_43765716746470_kernel(
    const float* __restrict__ x, float* __restrict__ out, int nGroups) {
  const int lane = threadIdx.x & 31;   // wave32 on gfx1250
  const int wave = threadIdx.x >> 5;
  const int g = blockIdx.x * (blockDim.x >> 5) + wave;
  if (g >= nGroups) return;

  const size_t gbase = (size_t)g * GROUP_SIZE;

  // Coalesced 16B/lane load: wave covers the full 512B group.
  float4 v = reinterpret_cast<const float4*>(x + gbase)[lane];

  // Per-lane min/max over 4 elements, then wave32 butterfly reduction.
  float vmin = fminf(fminf(v.x, v.y), fminf(v.z, v.w));
  float vmax = fmaxf(fmaxf(v.x, v.y), fmaxf(v.z, v.w));
#pragma unroll
  for (int m = 16; m >= 1; m >>= 1) {
    vmin = fminf(vmin, __shfl_xor(vmin, m, 32));
    vmax = fmaxf(vmax, __shfl_xor(vmax, m, 32));
  }

  // scale = clip((max-min)/15, 1e-6); zero = -min/scale; z_int = round_half_even(zero)
  float scale = (vmax - vmin) * (1.0f / (float)QMAX);
  scale = fmaxf(scale, CLIPMIN);
  const float inv_scale = 1.0f / scale;
  const float z_int = rintf(-vmin * inv_scale);

  // Constant terms of round_soft are bit-exact in fp32:
  //   tanh(1000*(k+0.5)) == 1.0f for all k (tanh(x)=1-2e^{-2x}; error << 2^-24)
  //   => sum_k tanh(1000*(k+0.5)) == 15.0f exactly
  //   denom = 2*tanh(1000) == 2.0f exactly
  const float base      = (float)QMAX;  // 15.0f
  const float inv_denom = 0.5f;

  float r[4] = {v.x, v.y, v.z, v.w};
#pragma unroll
  for (int i = 0; i < 4; ++i) {
    const float xfp = fmaf(r[i], inv_scale, z_int);  // x/scale + z_int
    float s = base;
#pragma unroll
    for (int k = 0; k < QMAX; ++k) {
      // T*(xfp-(k+0.5)) with T*(k+0.5)=500+1000k exactly representable:
      // single v_fma_f32 feeding v_tanh_f32 (TRANS co-executes with VALU).
      const float ck = 500.0f + 1000.0f * (float)k;
      s += dev_tanh(fmaf(T_SHARP, xfp, -ck));
    }
    const float x_int = s * inv_denom;               // round_soft(x_fp)
    r[i] = (x_int - z_int) * scale;                  // dequantize
  }

  float4 o;
  o.x = r[0]; o.y = r[1]; o.z = r[2]; o.w = r[3];
  reinterpret_cast<float4*>(out + gbase)[lane] = o;  // coalesced b128 store
}

extern "C" void kernel_launch(void* const* d_in, const int* in_sizes, int n_in,
                              void* d_out, int out_size, void* d_ws, size_t ws_size,
                              hipStream_t stream) {
  const float* x = (const float*)d_in[0];
  float* out = (float*)d_out;
  const int n = in_sizes[0];              // 2048*4096 = 8388608
  const int nGroups = n / GROUP_SIZE;     // 65536 groups of 128
  const int wavesPerBlock = 256 / 32;     // 8 groups per 256-thread block
  const int blocks = (nGroups + wavesPerBlock - 1) / wavesPerBlock;
  WeightQuantizer_43765716746470_kernel<<<blocks, 256, 0, stream>>>(x, out, nGroups);
}